// Decorrelation_10934986736132
// MI455X (gfx1250) — compile-verified
//
#include <hip/hip_runtime.h>

// Decorrelation (Bernstein-spline pair coupling) as a single fp32 WMMA GEMM:
//   wb[n, c*12+k] = C(10,k) x_c^k (1-x_c)^(10-k) * in[n,c]   (K padded 11->12, total 144)
//   Wt[v][c*12+k] = params[k, pair(v,c)] (c<v), 1.0 (c==v, partition of unity), else 0
//   out[n,v] = sum_kk wb[n,kk] * Wt[v][kk]     ==  in[n,v] + pair terms
// Each wave: 32 samples = two 16x16 accumulator tiles, 36 chained v_wmma_f32_16x16x4_f32 each.

#define D_        12
#define KDEG      11      // degree+1 coefficients per pair
#define KPAD      12      // padded per-variable K stride
#define KK        144     // 12 vars * 12 padded K
#define NSTEPS    36      // KK / 4 (WMMA K=4 per instruction)
#define ROWSTRIDE 148     // LDS row stride in floats: 16B-aligned, bank-conflict-free loads
#define WAVES     4
#define BLOCK     (WAVES * 32)
#define NPAIRS    66

typedef __attribute__((ext_vector_type(2))) float v2f;
typedef __attribute__((ext_vector_type(8))) float v8f;

__global__ __launch_bounds__(BLOCK) void decor_wmma_kernel(
    const float* __restrict__ inp,      // [N, 12]
    const float* __restrict__ params,   // [11, 66] row-major
    const float* __restrict__ prange,   // [2, 12]
    float* __restrict__ out,            // [N, 12]
    int n)
{
    extern __shared__ __align__(16) float smem[];
    float* Wt = smem;                         // [16][ROWSTRIDE]  (rows 12..15 stay zero)
    float* wb = smem + 16 * ROWSTRIDE;        // [BLOCK][ROWSTRIDE]

    const int tid  = threadIdx.x;
    const int lane = tid & 31;
    const int wave = tid >> 5;

    // ---- build Wt: zero, scatter params, fold identity ----
    for (int i = tid; i < 16 * ROWSTRIDE; i += BLOCK) Wt[i] = 0.0f;
    __syncthreads();
    for (int idx = tid; idx < KDEG * NPAIRS; idx += BLOCK) {
        int k = idx / NPAIRS;
        int p = idx % NPAIRS;                 // pair index, tril row-major: p = v(v-1)/2 + c
        int v = 1;
        while (((v + 1) * v) / 2 <= p) v++;
        int c = p - (v * (v - 1)) / 2;
        Wt[v * ROWSTRIDE + c * KPAD + k] = params[idx];
    }
    // identity fold: Bernstein basis sums to 1, so in[n,v] == sum_k wb[n, v*KPAD+k]
    for (int idx = tid; idx < D_ * KDEG; idx += BLOCK) {
        int v = idx / KDEG;
        int k = idx % KDEG;
        Wt[v * ROWSTRIDE + v * KPAD + k] = 1.0f;
    }
    __syncthreads();

    // ---- phase 1: weighted Bernstein basis for one sample per lane -> LDS ----
    long s = (long)blockIdx.x * BLOCK + tid;
    long sc = (s < n) ? s : (long)(n - 1);        // clamp: duplicate work, stores predicated
    const float4* row = (const float4*)(inp + sc * D_);
    float4 q0 = row[0], q1 = row[1], q2 = row[2];
    float xin[D_] = {q0.x, q0.y, q0.z, q0.w, q1.x, q1.y, q1.z, q1.w,
                     q2.x, q2.y, q2.z, q2.w};

    const float BIN[KDEG] = {1.f, 10.f, 45.f, 120.f, 210.f, 252.f,
                             210.f, 120.f, 45.f, 10.f, 1.f};
    float* myrow = wb + tid * ROWSTRIDE;
#pragma unroll
    for (int c = 0; c < D_; ++c) {
        float lowv  = prange[c];
        float highv = prange[D_ + c];
        float width = highv - lowv;
        float lo  = lowv - 0.1f * width;          // span-inflated lower bound
        float isc = 1.0f / (1.2f * width);        // 1 / (hi' - lo')
        float x = (xin[c] - lo) * isc;
        float y = 1.0f - x;
        float xp[KDEG], yp[KDEG];
        xp[0] = 1.0f; yp[0] = 1.0f;
#pragma unroll
        for (int k = 1; k < KDEG; ++k) { xp[k] = xp[k - 1] * x; yp[k] = yp[k - 1] * y; }
        float w = xin[c];
        float b[KPAD];
#pragma unroll
        for (int k = 0; k < KDEG; ++k) b[k] = BIN[k] * xp[k] * yp[10 - k] * w;
        b[11] = 0.0f;
        float4* dst = (float4*)(myrow + c * KPAD);
        dst[0] = make_float4(b[0], b[1], b[2],  b[3]);
        dst[1] = make_float4(b[4], b[5], b[6],  b[7]);
        dst[2] = make_float4(b[8], b[9], b[10], b[11]);
    }
    __syncthreads();

    // ---- phase 2: two 16x16 tiles per wave, 36 chained f32 WMMAs each ----
    const int half = lane >> 4;                   // selects K pair {0,1} vs {2,3}
    const int m    = lane & 15;                   // A: row index; B/D: column index
    const float* Arow0 = wb + (wave * 32 + m) * ROWSTRIDE + 2 * half;        // samples 0..15
    const float* Arow1 = Arow0 + 16 * ROWSTRIDE;                             // samples 16..31
    const float* Brow  = Wt + m * ROWSTRIDE + 2 * half;                      // column v = m

    v8f acc0 = {};
    v8f acc1 = {};
#pragma unroll
    for (int step = 0; step < NSTEPS; ++step) {
        v2f a0 = *(const v2f*)(Arow0 + 4 * step);
        v2f a1 = *(const v2f*)(Arow1 + 4 * step);
        v2f bb = *(const v2f*)(Brow  + 4 * step);
        acc0 = __builtin_amdgcn_wmma_f32_16x16x4_f32(
                   false, a0, false, bb, (short)0, acc0, false, false);
        acc1 = __builtin_amdgcn_wmma_f32_16x16x4_f32(
                   false, a1, false, bb, (short)0, acc1, false, false);
    }

    // ---- store: C/D layout VGPR r -> (M = r + 8*half, N = m) ----
    long tileBase = (long)blockIdx.x * BLOCK + wave * 32;
#pragma unroll
    for (int r = 0; r < 8; ++r) {
        long m0 = tileBase + r + half * 8;
        long m1 = m0 + 16;
        if (m < D_ && m0 < n) out[m0 * D_ + m] = acc0[r];
        if (m < D_ && m1 < n) out[m1 * D_ + m] = acc1[r];
    }
}

extern "C" void kernel_launch(void* const* d_in, const int* in_sizes, int n_in,
                              void* d_out, int out_size, void* d_ws, size_t ws_size,
                              hipStream_t stream) {
    const float* inp    = (const float*)d_in[0];
    const float* params = (const float*)d_in[1];
    const float* prange = (const float*)d_in[2];
    float* out = (float*)d_out;
    int n = in_sizes[0] / D_;
    int blocks = (n + BLOCK - 1) / BLOCK;
    size_t shmem = (size_t)(16 + BLOCK) * ROWSTRIDE * sizeof(float);  // ~85 KB
    decor_wmma_kernel<<<blocks, BLOCK, shmem, stream>>>(inp, params, prange, out, n);
}